// HoffmanSwarmV2_6373731467947
// MI455X (gfx1250) — compile-verified
//
#include <hip/hip_runtime.h>
#include <hip/hip_bf16.h>
#include <math.h>

// ---------------------------------------------------------------------------
// HoffmanSwarmV2 for MI455X (gfx1250): fp32 WMMA 16x16x4 path, wave32,
// with double-buffered async global->LDS staging of Gram-matrix tiles.
// ---------------------------------------------------------------------------

typedef __attribute__((ext_vector_type(2))) float v2f;
typedef __attribute__((ext_vector_type(8))) float v8f;

#define WMMA_F32(a, b, c) \
  __builtin_amdgcn_wmma_f32_16x16x4_f32(false, (a), false, (b), (short)0, (c), false, false)

// LDS tile layout: 16 rows x 64 floats, padded row stride 68 floats so that
// column reads (stride 68 dwords -> bank 4n+k mod 64) are conflict-free.
#define TROW 68
#define TILE_F (16 * TROW)

// A fragment: 16x4 fp32, row-major A with leading dim lda.
static __device__ __forceinline__ v2f load_a_frag(const float* A, int lda, int k0, int lane) {
  int m  = lane & 15;
  int kk = k0 + ((lane >> 4) << 1);
  v2f a;
  a.x = A[(size_t)m * lda + kk];
  a.y = A[(size_t)m * lda + kk + 1];
  return a;
}

// B fragment: 4x16 fp32 from row-major KxN matrix with leading dim ldb.
static __device__ __forceinline__ v2f load_b_frag(const float* Bm, int ldb, int k0, int lane) {
  int n  = lane & 15;
  int kk = k0 + ((lane >> 4) << 1);
  v2f b;
  b.x = Bm[(size_t)kk * ldb + n];
  b.y = Bm[(size_t)(kk + 1) * ldb + n];
  return b;
}

// B fragment for Gram matmuls from an LDS-staged tile (16 rows x 64, stride TROW).
// B[k][n] = tile[n][k].
static __device__ __forceinline__ v2f load_bt_lds(const float* kt, int k0, int lane) {
  int n  = lane & 15;
  int kk = k0 + ((lane >> 4) << 1);
  v2f b;
  b.x = kt[n * TROW + kk];
  b.y = kt[n * TROW + kk + 1];
  return b;
}

// Async DMA of one 16x64 fp32 tile (row-major, ld=64) into padded LDS layout.
// 8 x global_load_async_to_lds_b128: each lane moves 16B, fully coalesced on
// the global side, scattered into the padded layout on the LDS side.
static __device__ __forceinline__ void async_tile_load(const float* __restrict__ gsrc,
                                                       unsigned lds_base, int lane) {
#pragma unroll
  for (int i = 0; i < 8; ++i) {
    int e = i * 128 + lane * 4;              // float index within tile
    int r = e >> 6, c = e & 63;
    unsigned lds_off = lds_base + (unsigned)((r * TROW + c) * 4);
    const float* gp = gsrc + e;
    asm volatile("global_load_async_to_lds_b128 %0, %1, off"
                 :: "v"(lds_off), "v"(gp) : "memory");
  }
}

// ---------------------------------------------------------------------------
// Kernel 1: Q/K/C projections. grid = (BN/16, 3), block = 32 (one wave).
// ---------------------------------------------------------------------------
__global__ void k_proj(const float* __restrict__ act,
                       const float* __restrict__ Wq, const float* __restrict__ bq,
                       const float* __restrict__ Wk, const float* __restrict__ bk,
                       const float* __restrict__ Wc, const float* __restrict__ bc,
                       float* __restrict__ Qp, float* __restrict__ Kp,
                       float* __restrict__ Cn) {
  const int D = 256, P = 64;
  const int rt   = blockIdx.x;
  const int wt   = blockIdx.y;     // 0=Q, 1=K, 2=C
  const int lane = threadIdx.x;

  const float* W    = (wt == 0) ? Wq : (wt == 1) ? Wk : Wc;
  const float* bias = (wt == 0) ? bq : (wt == 1) ? bk : bc;
  float*       Out  = (wt == 0) ? Qp : (wt == 1) ? Kp : Cn;
  const float* A    = act + (size_t)rt * 16 * D;

  __shared__ float s_tile[16][65];
  __shared__ float s_invn[16];

  for (int ct = 0; ct < 4; ++ct) {
    v8f c = {};
    for (int k0 = 0; k0 < D; k0 += 4) {
      v2f a = load_a_frag(A, D, k0, lane);
      v2f b = load_b_frag(W + ct * 16, P, k0, lane);
      c = WMMA_F32(a, b, c);
    }
    const int col   = lane & 15;
    const int rbase = (lane >> 4) << 3;
    const float bv  = bias[ct * 16 + col];
#pragma unroll
    for (int r = 0; r < 8; ++r) s_tile[rbase + r][ct * 16 + col] = c[r] + bv;
  }
  __syncthreads();

  if (wt == 2 && lane < 16) {
    float s = 0.f;
#pragma unroll 8
    for (int c2 = 0; c2 < 64; ++c2) { float v = s_tile[lane][c2]; s += v * v; }
    s_invn[lane] = 1.f / fmaxf(sqrtf(s), 1e-12f);
  }
  __syncthreads();

  for (int i = lane; i < 16 * 64; i += 32) {
    int r = i >> 6, cc = i & 63;
    float v = s_tile[r][cc];
    if (wt == 2) v *= s_invn[r];
    Out[((size_t)rt * 16 + r) * 64 + cc] = v;
  }
}

// ---------------------------------------------------------------------------
// Kernel 2: fused scores -> top-8 -> softmax -> neighbor gather (incoming).
// Streams 256 16x16 score tiles; K-tiles double-buffered via async->LDS DMA.
// grid = B * (N/16), block = 32.
// ---------------------------------------------------------------------------
__global__ void k_score(const float* __restrict__ Qp, const float* __restrict__ Kp,
                        const float* __restrict__ states,
                        float* __restrict__ incoming, int N) {
  const int P = 64, D = 256;
  const int bi   = blockIdx.x;
  const int b    = bi / (N / 16);
  const int rt   = bi % (N / 16);
  const int lane = threadIdx.x;
  const int NT   = N / 16;

  const float* Arow = Qp + ((size_t)b * N + (size_t)rt * 16) * P;
  const float* Kb   = Kp + (size_t)b * N * P;

  __shared__ float s_kt[2][TILE_F];
  __shared__ float s_stage[16][17];
  __shared__ float s_w[16][8];
  __shared__ int   s_i[16][8];

  const unsigned ldsb0 = (unsigned)(uintptr_t)&s_kt[0][0];
  const unsigned ldsb1 = (unsigned)(uintptr_t)&s_kt[1][0];

  v2f afr[16];
#pragma unroll
  for (int t = 0; t < 16; ++t) afr[t] = load_a_frag(Arow, P, t * 4, lane);

  float tv[8]; int ti[8];
#pragma unroll
  for (int t = 0; t < 8; ++t) { tv[t] = -1e30f; ti[t] = 0; }

  async_tile_load(Kb, ldsb0, lane);

  for (int ct = 0; ct < NT; ++ct) {
    const int cur = ct & 1;
    if (ct + 1 < NT) {
      async_tile_load(Kb + (size_t)(ct + 1) * 16 * P, cur ? ldsb0 : ldsb1, lane);
      asm volatile("s_wait_asynccnt 0x8" ::: "memory");
    } else {
      asm volatile("s_wait_asynccnt 0x0" ::: "memory");
    }

    const float* kt = &s_kt[cur][0];
    v8f c = {};
#pragma unroll
    for (int t = 0; t < 16; ++t) {
      v2f bfr = load_bt_lds(kt, t * 4, lane);
      c = WMMA_F32(afr[t], bfr, c);
    }
    const int col   = lane & 15;
    const int rbase = (lane >> 4) << 3;
#pragma unroll
    for (int r = 0; r < 8; ++r) s_stage[rbase + r][col] = c[r] * 0.125f; // 1/sqrt(64)
    __syncthreads();

    if (lane < 16) {
#pragma unroll 4
      for (int t = 0; t < 16; ++t) {
        float v = s_stage[lane][t];
        int mn = 0;
#pragma unroll
        for (int q = 1; q < 8; ++q) if (tv[q] < tv[mn]) mn = q;
        if (v > tv[mn]) { tv[mn] = v; ti[mn] = ct * 16 + t; }
      }
    }
    __syncthreads();
  }

  // softmax over the surviving 8 (masked entries underflow to exactly 0)
  if (lane < 16) {
    float mx = tv[0];
#pragma unroll
    for (int q = 1; q < 8; ++q) mx = fmaxf(mx, tv[q]);
    float s = 0.f, e[8];
#pragma unroll
    for (int q = 0; q < 8; ++q) { e[q] = __expf(tv[q] - mx); s += e[q]; }
    float inv = 1.f / s;
#pragma unroll
    for (int q = 0; q < 8; ++q) { s_w[lane][q] = e[q] * inv; s_i[lane][q] = ti[q]; }
  }
  __syncthreads();

  // incoming[row] = sum_t w_t * states[b, idx_t, :]
  for (int j = 0; j < 16; ++j) {
    float acc[8];
#pragma unroll
    for (int d = 0; d < 8; ++d) acc[d] = 0.f;
#pragma unroll
    for (int t = 0; t < 8; ++t) {
      const float w = s_w[j][t];
      const float* src = states + ((size_t)b * N + s_i[j][t]) * D;
#pragma unroll
      for (int d = 0; d < 8; ++d) acc[d] += w * src[lane + d * 32];
    }
    float* dst = incoming + ((size_t)b * N + (size_t)rt * 16 + j) * D;
#pragma unroll
    for (int d = 0; d < 8; ++d) dst[lane + d * 32] = acc[d];
  }
}

// ---------------------------------------------------------------------------
// Kernel 3: coalition mix. Streams sim tiles (WMMA on normalized C proj,
// double-buffered async->LDS), ballots entries > 0.7, accumulates hit rows.
// out_states = 0.8*incoming + 0.2*combined. grid = B*(N/16), block = 32.
// ---------------------------------------------------------------------------
__global__ void k_coalition(const float* __restrict__ Cn,
                            const float* __restrict__ incoming,
                            float* __restrict__ out_states, int N) {
  const int P = 64, D = 256;
  const int bi   = blockIdx.x;
  const int b    = bi / (N / 16);
  const int rt   = bi % (N / 16);
  const int lane = threadIdx.x;
  const int NT   = N / 16;

  const float* Arow = Cn + ((size_t)b * N + (size_t)rt * 16) * P;
  const float* Cb   = Cn + (size_t)b * N * P;

  __shared__ float s_acc[16][256];
  __shared__ float s_kt[2][TILE_F];
  __shared__ float s_stage[16][17];

  const unsigned ldsb0 = (unsigned)(uintptr_t)&s_kt[0][0];
  const unsigned ldsb1 = (unsigned)(uintptr_t)&s_kt[1][0];

  float* sp = &s_acc[0][0];
  for (int i = lane; i < 16 * 256; i += 32) sp[i] = 0.f;

  v2f afr[16];
#pragma unroll
  for (int t = 0; t < 16; ++t) afr[t] = load_a_frag(Arow, P, t * 4, lane);

  float cnt[16];
#pragma unroll
  for (int j = 0; j < 16; ++j) cnt[j] = 0.f;

  __syncthreads();

  async_tile_load(Cb, ldsb0, lane);

  for (int ct = 0; ct < NT; ++ct) {
    const int cur = ct & 1;
    if (ct + 1 < NT) {
      async_tile_load(Cb + (size_t)(ct + 1) * 16 * P, cur ? ldsb0 : ldsb1, lane);
      asm volatile("s_wait_asynccnt 0x8" ::: "memory");
    } else {
      asm volatile("s_wait_asynccnt 0x0" ::: "memory");
    }

    const float* kt = &s_kt[cur][0];
    v8f c = {};
#pragma unroll
    for (int t = 0; t < 16; ++t) {
      v2f bfr = load_bt_lds(kt, t * 4, lane);
      c = WMMA_F32(afr[t], bfr, c);
    }
    const int col   = lane & 15;
    const int rbase = (lane >> 4) << 3;
#pragma unroll
    for (int r = 0; r < 8; ++r) s_stage[rbase + r][col] = c[r];
    __syncthreads();

    for (int j = 0; j < 16; ++j) {
      bool p = (lane < 16) && (s_stage[j][lane] > 0.7f);
      unsigned long long m = __ballot(p) & 0xFFFFull;
      cnt[j] += (float)__popcll(m);
      while (m) {
        int cc = __ffsll((long long)m) - 1;
        m &= m - 1;
        const float* src = incoming + ((size_t)b * N + (size_t)ct * 16 + cc) * D;
#pragma unroll
        for (int d = 0; d < 8; ++d) s_acc[j][lane + d * 32] += src[lane + d * 32];
      }
    }
    __syncthreads();
  }

  for (int j = 0; j < 16; ++j) {
    const float ic = 1.f / (cnt[j] + 1e-8f);
    const size_t row = (size_t)b * N + (size_t)rt * 16 + j;
    const float* inc = incoming + row * D;
    float* dst = out_states + row * D;
#pragma unroll
    for (int d = 0; d < 8; ++d) {
      int idx = lane + d * 32;
      dst[idx] = 0.8f * inc[idx] + 0.2f * (s_acc[j][idx] * ic);
    }
  }
}

// ---------------------------------------------------------------------------
// Kernel 4: HDC projection + memory write + cosine read, fully fused.
// grid = B*(N/16), block = 32.
// ---------------------------------------------------------------------------
__global__ void k_hdc(const float* __restrict__ out_states,
                      const float* __restrict__ Wh, const float* __restrict__ bh,
                      const float* __restrict__ keys, const float* __restrict__ pos_codes,
                      const int* __restrict__ stepp,
                      const float* __restrict__ mem,
                      float* __restrict__ new_mem, float* __restrict__ mem_sim, int N) {
  const int D = 256, H = 1024, NC = 256;
  const int bi   = blockIdx.x;
  const int b    = bi / (N / 16);
  const int rt   = bi % (N / 16);
  const int lane = threadIdx.x;

  const int step = stepp[0];
  const float* pos = pos_codes + (size_t)(((step % NC) + NC) % NC) * H;
  const float* A = out_states + ((size_t)b * N + (size_t)rt * 16) * D;

  __shared__ float s_red[3][16][16];

  float dotA[8], naA[8], nbA[8];
#pragma unroll
  for (int r = 0; r < 8; ++r) { dotA[r] = 0.f; naA[r] = 0.f; nbA[r] = 0.f; }

  for (int ct = 0; ct < H / 16; ++ct) {
    v8f c = {};
    for (int k0 = 0; k0 < D; k0 += 4) {
      v2f a  = load_a_frag(A, D, k0, lane);
      v2f bf = load_b_frag(Wh + ct * 16, H, k0, lane);
      c = WMMA_F32(a, bf, c);
    }
    const int col   = ct * 16 + (lane & 15);
    const int rbase = (lane >> 4) << 3;
    const float posv = pos[col];
    const float bv   = bh[col];
#pragma unroll
    for (int r = 0; r < 8; ++r) {
      const int row = rbase + r;
      const int n   = rt * 16 + row;
      const size_t gi = ((size_t)b * N + n) * H + col;
      float h   = tanhf(3.f * (c[r] + bv));
      float per = h * keys[(size_t)n * H + col];
      float ne  = 0.95f * mem[gi] + 0.05f * (per * posv);
      new_mem[gi] = ne;
      dotA[r] += ne * per;
      naA[r]  += ne * ne;
      nbA[r]  += per * per;
    }
  }

  const int col   = lane & 15;
  const int rbase = (lane >> 4) << 3;
#pragma unroll
  for (int r = 0; r < 8; ++r) {
    s_red[0][rbase + r][col] = dotA[r];
    s_red[1][rbase + r][col] = naA[r];
    s_red[2][rbase + r][col] = nbA[r];
  }
  __syncthreads();

  if (lane < 16) {
    float dsum = 0.f, nasum = 0.f, nbsum = 0.f;
#pragma unroll 4
    for (int c2 = 0; c2 < 16; ++c2) {
      dsum  += s_red[0][lane][c2];
      nasum += s_red[1][lane][c2];
      nbsum += s_red[2][lane][c2];
    }
    float na = fmaxf(sqrtf(nasum), 1e-8f);
    float nb = fmaxf(sqrtf(nbsum), 1e-8f);
    mem_sim[(size_t)b * N + (size_t)rt * 16 + lane] = dsum / (na * nb);
  }
}

// ---------------------------------------------------------------------------
// Host launcher
// ---------------------------------------------------------------------------
extern "C" void kernel_launch(void* const* d_in, const int* in_sizes, int n_in,
                              void* d_out, int out_size, void* d_ws, size_t ws_size,
                              hipStream_t stream) {
  const int B = 4, N = 4096, D = 256, P = 64, H = 1024;
  const int BN = B * N;

  const float* agent_states  = (const float*)d_in[0];
  const float* agent_actions = (const float*)d_in[1];
  const float* episodic_mem  = (const float*)d_in[2];
  const float* Wq = (const float*)d_in[3];  const float* bq = (const float*)d_in[4];
  const float* Wk = (const float*)d_in[5];  const float* bk = (const float*)d_in[6];
  const float* Wc = (const float*)d_in[7];  const float* bc = (const float*)d_in[8];
  const float* Wh = (const float*)d_in[9];  const float* bh = (const float*)d_in[10];
  const float* agent_keys = (const float*)d_in[11];
  const float* pos_codes  = (const float*)d_in[12];
  const int*   step       = (const int*)d_in[13];

  float* out_states  = (float*)d_out;                                   // B*N*D
  float* new_mem     = (float*)d_out + (size_t)BN * D;                  // B*N*H
  float* mem_sim     = (float*)d_out + (size_t)BN * D + (size_t)BN * H; // B*N

  float* ws       = (float*)d_ws;
  float* Qp       = ws;                                  // BN*P
  float* Kp       = ws + (size_t)BN * P;                 // BN*P
  float* Cn       = ws + (size_t)BN * P * 2;             // BN*P
  float* incoming = ws + (size_t)BN * P * 3;             // BN*D

  dim3 blk(32, 1, 1);

  k_proj<<<dim3(BN / 16, 3, 1), blk, 0, stream>>>(
      agent_actions, Wq, bq, Wk, bk, Wc, bc, Qp, Kp, Cn);

  k_score<<<dim3(B * (N / 16), 1, 1), blk, 0, stream>>>(
      Qp, Kp, agent_states, incoming, N);

  k_coalition<<<dim3(B * (N / 16), 1, 1), blk, 0, stream>>>(
      Cn, incoming, out_states, N);

  k_hdc<<<dim3(B * (N / 16), 1, 1), blk, 0, stream>>>(
      out_states, Wh, bh, agent_keys, pos_codes, step,
      episodic_mem, new_mem, mem_sim, N);
}